// Pooling_Block_90082644066733
// MI455X (gfx1250) — compile-verified
//
#include <hip/hip_runtime.h>

#define B_     4
#define N_     8192
#define NP_    2048
#define NS_    32
#define CIN_   128
#define COUT_  256
#define RAD2   0.04f      // RADIUS^2
#define ALPHA_ 0.2f
#define EPS_   1e-5f

typedef __attribute__((ext_vector_type(2))) float v2f;
typedef __attribute__((ext_vector_type(8))) float v8f;

// ---------------------------------------------------------------------------
// Kernel 1: furthest point sampling. One block per batch; points live in VGPRs.
// ---------------------------------------------------------------------------
#define FPS_T  512
#define FPS_PP (N_ / FPS_T)   // 16 points per thread

__global__ __launch_bounds__(FPS_T)
void fps_kernel(const float* __restrict__ xyz, int* __restrict__ fidx) {
  __shared__ float swv[FPS_T / 32];
  __shared__ int   swi[FPS_T / 32];
  __shared__ int   s_cur;

  const int b    = blockIdx.x;
  const int tid  = threadIdx.x;
  const int lane = tid & 31;
  const int wv   = tid >> 5;
  const float* p = xyz + (size_t)b * N_ * 3;

  float px[FPS_PP], py[FPS_PP], pz[FPS_PP], md[FPS_PP];
  const int i0 = tid * FPS_PP;
#pragma unroll
  for (int k = 0; k < FPS_PP; ++k) {
    px[k] = p[(i0 + k) * 3 + 0];
    py[k] = p[(i0 + k) * 3 + 1];
    pz[k] = p[(i0 + k) * 3 + 2];
    md[k] = 3.4e38f;                 // min(inf, d(.,pt0)) == reference init
  }
  if (tid == 0) fidx[(size_t)b * NP_] = 0;
  int cur = 0;

  for (int it = 1; it < NP_; ++it) {
    const float cx = p[cur * 3 + 0];
    const float cy = p[cur * 3 + 1];
    const float cz = p[cur * 3 + 2];
    // fused min-dist update + local argmax
    float bv = -1.f; int bi = 0x7fffffff;
#pragma unroll
    for (int k = 0; k < FPS_PP; ++k) {
      const float dx = px[k] - cx, dy = py[k] - cy, dz = pz[k] - cz;
      const float d  = dx * dx + dy * dy + dz * dz;
      const float v  = md[k] < d ? md[k] : d;
      md[k] = v;
      if (v > bv) { bv = v; bi = i0 + k; }   // ascending k => first-max kept
    }
    // wave32 argmax reduce (tie -> lowest index, matching jnp.argmax)
#pragma unroll
    for (int off = 16; off > 0; off >>= 1) {
      const float ov = __shfl_xor(bv, off);
      const int   oi = __shfl_xor(bi, off);
      if (ov > bv || (ov == bv && oi < bi)) { bv = ov; bi = oi; }
    }
    if (lane == 0) { swv[wv] = bv; swi[wv] = bi; }
    __syncthreads();
    if (wv == 0) {
      const int nw = FPS_T / 32;
      bv = lane < nw ? swv[lane] : -1.f;
      bi = lane < nw ? swi[lane] : 0x7fffffff;
#pragma unroll
      for (int off = 16; off > 0; off >>= 1) {
        const float ov = __shfl_xor(bv, off);
        const int   oi = __shfl_xor(bi, off);
        if (ov > bv || (ov == bv && oi < bi)) { bv = ov; bi = oi; }
      }
      if (lane == 0) { s_cur = bi; fidx[(size_t)b * NP_ + it] = bi; }
    }
    __syncthreads();
    cur = s_cur;
  }
}

// ---------------------------------------------------------------------------
// Kernel 2: ball query (first NS in index order) + feature gather + max pool.
// One wave32 per query point; NS == 32 == wavefront, slot s lives on lane s.
// Pooled output is stored k-pair interleaved for b64 WMMA B-fragment loads:
//   pooled[((b*64 + c/2)*NP + n)*2 + (c&1)]
// ---------------------------------------------------------------------------
__global__ __launch_bounds__(256)
void ballq_pool_kernel(const float* __restrict__ xyz,
                       const float* __restrict__ feats,
                       const int*   __restrict__ fidx,
                       float*       __restrict__ pooled) {
  const int gw   = blockIdx.x * (blockDim.x >> 5) + (threadIdx.x >> 5);
  const int lane = threadIdx.x & 31;
  const int b    = gw / NP_;
  const int j    = gw % NP_;

  const float* p = xyz + (size_t)b * N_ * 3;
  const int   fid = fidx[(size_t)b * NP_ + j];
  const float qx = p[fid * 3 + 0], qy = p[fid * 3 + 1], qz = p[fid * 3 + 2];

  int cnt = 0;     // uniform across wave
  int nbr = 0;     // neighbor index owned by slot == lane
  for (int base = 0; base < N_ && cnt < NS_; base += 32) {
    const int i  = base + lane;
    const float dx = p[i * 3 + 0] - qx;
    const float dy = p[i * 3 + 1] - qy;
    const float dz = p[i * 3 + 2] - qz;
    const float d2 = dx * dx + dy * dy + dz * dz;
    const unsigned mask = (unsigned)__ballot(d2 < RAD2);   // wave32: 32-bit mask
    const int pc = __popc(mask);
    const int r  = lane - cnt;           // lane claims slot==lane
    if (r >= 0 && r < pc) {
      unsigned m = mask;
      for (int t = 0; t < r; ++t) m &= m - 1;  // drop r lowest set bits
      nbr = base + (__ffs(m) - 1);
    }
    cnt += pc;
    if (cnt > NS_) cnt = NS_;
  }
  const int first = __shfl(nbr, 0);
  if (lane >= cnt) nbr = (cnt > 0) ? first : 0;   // reference fallback semantics

  const float* fb = feats + (size_t)b * CIN_ * N_;
  // interleaved pooled base for (b, j)
  float* pout = pooled + (size_t)b * (CIN_ / 2) * NP_ * 2 + (size_t)j * 2;
#pragma unroll 4
  for (int c = 0; c < CIN_; ++c) {
    float f = fb[(size_t)c * N_ + nbr];
#pragma unroll
    for (int off = 16; off > 0; off >>= 1) f = fmaxf(f, __shfl_xor(f, off));
    if (lane == 0) pout[(size_t)(c >> 1) * NP_ * 2 + (c & 1)] = f;
  }
}

// ---------------------------------------------------------------------------
// Kernel 3: out = W(256x128) x pooled(128x2048) per batch via
// v_wmma_f32_16x16x4_f32. Each wave computes a 16(M) x 64(N) strip with 4
// accumulators, reusing one A fragment across 4 WMMAs. Fused BN + leaky ReLU.
// Grid exactly covers all strips -> EXEC all 1s (WMMA requirement).
// ---------------------------------------------------------------------------
__global__ __launch_bounds__(256)
void gemm_bn_relu_kernel(const float* __restrict__ W,
                         const float* __restrict__ pooled,
                         const float* __restrict__ gamma,
                         const float* __restrict__ beta,
                         const float* __restrict__ rmean,
                         const float* __restrict__ rvar,
                         float*       __restrict__ out) {
  const int gw   = blockIdx.x * (blockDim.x >> 5) + (threadIdx.x >> 5);
  const int lane = threadIdx.x & 31;
  const int stripsPerBatch = (COUT_ / 16) * (NP_ / 64);   // 16 * 32
  const int b  = gw / stripsPerBatch;
  const int t  = gw % stripsPerBatch;
  const int tm = t / (NP_ / 64);       // 0..15  (M tile)
  const int tn = t % (NP_ / 64);       // 0..31  (64-wide N strip)

  const int lmod = lane & 15;          // M (for A) / N (for B,D)
  const int lhi  = lane >> 4;          // hi half holds K=2,3 (A/B), M+8 (D)

  const int m  = tm * 16 + lmod;       // A row
  const int nn = tn * 64 + lmod;       // first B col of strip
  const float* Wrow = W + (size_t)m * CIN_ + 2 * lhi;               // 8B aligned
  // interleaved pooled: element (b, k, n) at ((b*64 + k/2)*NP + n)*2 + (k&1)
  const float* Bbase =
      pooled + ((size_t)b * (CIN_ / 2) + lhi) * NP_ * 2 + (size_t)nn * 2;

  v8f acc0 = {}, acc1 = {}, acc2 = {}, acc3 = {};
  for (int k0 = 0; k0 < CIN_; k0 += 4) {
    const v2f a = *(const v2f*)(Wrow + k0);                         // A[m][k0+2*lhi .. +1]
    const float* bp = Bbase + (size_t)(k0 >> 1) * NP_ * 2;
    const v2f b0 = *(const v2f*)(bp + 0 * 32);                      // B[k..k+1][nn+ 0]
    const v2f b1 = *(const v2f*)(bp + 1 * 32);                      // B[k..k+1][nn+16]
    const v2f b2 = *(const v2f*)(bp + 2 * 32);                      // B[k..k+1][nn+32]
    const v2f b3 = *(const v2f*)(bp + 3 * 32);                      // B[k..k+1][nn+48]
    acc0 = __builtin_amdgcn_wmma_f32_16x16x4_f32(false, a, false, b0, (short)0, acc0, false, false);
    acc1 = __builtin_amdgcn_wmma_f32_16x16x4_f32(false, a, false, b1, (short)0, acc1, false, false);
    acc2 = __builtin_amdgcn_wmma_f32_16x16x4_f32(false, a, false, b2, (short)0, acc2, false, false);
    acc3 = __builtin_amdgcn_wmma_f32_16x16x4_f32(false, a, false, b3, (short)0, acc3, false, false);
  }

  // D layout: VGPR r -> M = tm*16 + r + 8*lhi, N = col
#pragma unroll
  for (int r = 0; r < 8; ++r) {
    const int   o    = tm * 16 + r + 8 * lhi;
    const float inv  = gamma[o] * rsqrtf(rvar[o] + EPS_);
    const float bias = beta[o] - rmean[o] * inv;
    float* orow = out + ((size_t)b * COUT_ + o) * NP_ + nn;
    float v;
    v = acc0[r] * inv + bias; orow[ 0] = v > 0.f ? v : ALPHA_ * v;
    v = acc1[r] * inv + bias; orow[16] = v > 0.f ? v : ALPHA_ * v;
    v = acc2[r] * inv + bias; orow[32] = v > 0.f ? v : ALPHA_ * v;
    v = acc3[r] * inv + bias; orow[48] = v > 0.f ? v : ALPHA_ * v;
  }
}

// ---------------------------------------------------------------------------
extern "C" void kernel_launch(void* const* d_in, const int* in_sizes, int n_in,
                              void* d_out, int out_size, void* d_ws, size_t ws_size,
                              hipStream_t stream) {
  const float* xyz    = (const float*)d_in[0];
  const float* feats  = (const float*)d_in[1];
  const float* conv_w = (const float*)d_in[2];
  const float* gamma  = (const float*)d_in[3];
  const float* beta   = (const float*)d_in[4];
  const float* rmean  = (const float*)d_in[5];
  const float* rvar   = (const float*)d_in[6];
  float* out = (float*)d_out;

  int*   fidx   = (int*)d_ws;                                   // B*NP ints (32 KB)
  float* pooled = (float*)((char*)d_ws + (64u << 10));          // B*CIN*NP floats (4 MB)

  fps_kernel<<<B_, FPS_T, 0, stream>>>(xyz, fidx);

  ballq_pool_kernel<<<(B_ * NP_) / 8, 256, 0, stream>>>(xyz, feats, fidx, pooled);

  const int strips = B_ * (COUT_ / 16) * (NP_ / 64);            // 2048 waves
  gemm_bn_relu_kernel<<<strips / 8, 256, 0, stream>>>(conv_w, pooled, gamma, beta,
                                                      rmean, rvar, out);
}